// ConvSepKanCell_60198261621024
// MI455X (gfx1250) — compile-verified
//
#include <hip/hip_runtime.h>
#include <cstdint>

#define IN_C    16
#define OUT_C   16
#define COEF_M  11
#define HW_STR  16384        // 128*128 pixels per batch-slice
#define W_SIZE  3328
#define UW_OFF  2816
#define RW_OFF  3072

// ---- CDNA5 async global->LDS (ASYNCcnt-tracked) --------------------------
// GVS form: copies 4 bytes from the scalar base sbase plus the offset in voff to the LDS address in lds_off
__device__ __forceinline__ void aload(uint32_t lds_off, uint32_t voff, uint64_t sbase) {
  asm volatile("global_load_async_to_lds_b32 %0, %1, %2"
               :: "v"(lds_off), "v"(voff), "s"(sbase) : "memory");
}
template <int N>
__device__ __forceinline__ void wait_async() {
  asm volatile("s_wait_asynccnt %0" :: "n"(N) : "memory");
}
__device__ __forceinline__ void wait_ds0() {
  asm volatile("s_wait_dscnt 0" ::: "memory");
}

// Stage = 13 coalesced b32 async loads: 11 coef + uw + rw for one (i,o).
__device__ __forceinline__ void prefetch_stage(int k, int wv, uint32_t lds0,
                                               uint32_t hw4, uint64_t sbase) {
  const int ci = k >> 4;
  const int o  = k & 15;
  const int i  = 2 * wv + ci;
  const uint32_t l  = lds0 + (uint32_t)(k % 3) * (13u * 32u * 4u);
  const uint32_t vo = (uint32_t)(i * 176 + o * 11) * 65536u + hw4;
#pragma unroll
  for (int m = 0; m < 11; ++m)
    aload(l + (uint32_t)m * 128u, vo + (uint32_t)m * 65536u, sbase);
  aload(l + 11u * 128u, (uint32_t)(UW_OFF + i * 16 + o) * 65536u + hw4, sbase);
  aload(l + 12u * 128u, (uint32_t)(RW_OFF + i * 16 + o) * 65536u + hw4, sbase);
}

__global__ __launch_bounds__(256)
void kan_pixelwise_kernel(const float* __restrict__ xg,
                          const float* __restrict__ wg,
                          float* __restrict__ out) {
  // wave-private triple-buffered staging + cross-wave reduction buffer
  __shared__ float stage[8][3][13][32];   // 39936 B
  __shared__ float red[8][OUT_C][32];     // 16384 B

  const int lane = threadIdx.x & 31;
  const int wv   = threadIdx.x >> 5;
  const int p    = blockIdx.x * 32 + lane;   // global pixel id
  const int b    = p >> 14;                  // batch (uniform per block)
  const int hw   = p & (HW_STR - 1);

  const uint64_t sbase = (uint64_t)(uintptr_t)wg +
                         (uint64_t)b * ((uint64_t)W_SIZE * HW_STR * 4u);
  const uint32_t hw4  = (uint32_t)hw * 4u;
  const uint32_t lds0 = (uint32_t)(uintptr_t)(&stage[wv][0][0][lane]);

  // ---- cubic B-spline bases (Cox-de Boor, grid -1.75..1.75 step .25) ----
  float bas[2][COEF_M];
  float sil[2];
#pragma unroll
  for (int ci = 0; ci < 2; ++ci) {
    const int i = 2 * wv + ci;
    const float xv = xg[(((size_t)b * IN_C + i) << 14) + hw];
    float bs[14];
#pragma unroll
    for (int t = 0; t < 14; ++t) {
      const float g0 = -1.75f + 0.25f * (float)t;
      const float g1 = -1.75f + 0.25f * (float)(t + 1);
      bs[t] = (xv >= g0 && xv < g1) ? 1.0f : 0.0f;      // half-open, as reference
    }
#pragma unroll
    for (int pd = 1; pd <= 3; ++pd) {
#pragma unroll
      for (int t = 0; t < 14 - pd; ++t) {
        const float gt   = -1.75f + 0.25f * (float)t;
        const float gt1  = -1.75f + 0.25f * (float)(t + 1);
        const float gtp  = -1.75f + 0.25f * (float)(t + pd);
        const float gtp1 = -1.75f + 0.25f * (float)(t + pd + 1);
        bs[t] = (xv - gt) / (gtp - gt) * bs[t] +
                (gtp1 - xv) / (gtp1 - gt1) * bs[t + 1];
      }
    }
#pragma unroll
    for (int m = 0; m < COEF_M; ++m) bas[ci][m] = bs[m];
    sil[ci] = xv / (1.0f + expf(-xv));                  // silu
  }

  float acc[OUT_C];
#pragma unroll
  for (int o = 0; o < OUT_C; ++o) acc[o] = 0.0f;

  // ---- software pipeline: 3-deep async staging over 32 stages -----------
  prefetch_stage(0, wv, lds0, hw4, sbase);
  prefetch_stage(1, wv, lds0, hw4, sbase);

#pragma unroll
  for (int ci = 0; ci < 2; ++ci) {
#pragma unroll
    for (int o = 0; o < OUT_C; ++o) {
      const int k = ci * 16 + o;
      if (k < 30) {
        wait_ds0();                                   // WAR: buffer reuse
        prefetch_stage(k + 2, wv, lds0, hw4, sbase);
        wait_async<26>();                             // stage k complete
      } else if (k == 30) {
        wait_async<13>();
      } else {
        wait_async<0>();
      }
      const float* sb = &stage[wv][k % 3][0][lane];
      float sp = 0.0f;
#pragma unroll
      for (int m = 0; m < COEF_M; ++m) sp += sb[m * 32] * bas[ci][m];
      acc[o] += sb[11 * 32] * sp + sb[12 * 32] * sil[ci];
    }
  }

  // ---- deterministic cross-wave reduction over the channel split --------
#pragma unroll
  for (int o = 0; o < OUT_C; ++o) red[wv][o][lane] = acc[o];
  __syncthreads();
#pragma unroll
  for (int oi = 0; oi < 2; ++oi) {
    const int o = 2 * wv + oi;
    float s = 0.0f;
#pragma unroll
    for (int w2 = 0; w2 < 8; ++w2) s += red[w2][o][lane];
    out[(((size_t)b * OUT_C + o) << 14) + hw] = s;    // coalesced 128B store
  }
}

extern "C" void kernel_launch(void* const* d_in, const int* in_sizes, int n_in,
                              void* d_out, int out_size, void* d_ws, size_t ws_size,
                              hipStream_t stream) {
  (void)in_sizes; (void)n_in; (void)d_ws; (void)ws_size; (void)out_size;
  const float* x = (const float*)d_in[0];   // (2,16,128,128) f32
  const float* w = (const float*)d_in[1];   // (2,3328,128,128) f32
  float* out = (float*)d_out;               // (2,16,128,128) f32
  const int n_pix = 2 * 128 * 128;          // 32768
  dim3 grid(n_pix / 32), block(256);
  hipLaunchKernelGGL(kan_pixelwise_kernel, grid, block, 0, stream, x, w, out);
}